// GlobalAttention_9775345566244
// MI455X (gfx1250) — compile-verified
//
#include <hip/hip_runtime.h>
#include <hip/hip_bf16.h>
#include <math.h>

typedef __attribute__((ext_vector_type(2))) float v2f;
typedef __attribute__((ext_vector_type(4))) float v4f;
typedef __attribute__((ext_vector_type(8))) float v8f;

#define B_    32
#define C_    256
#define HW_   4096
#define A_    4
#define APAD_ 16      // key rows padded to WMMA N=16
#define MID_  1024
#define CA_   1024

// ---------------------------------------------------------------------------
// Kernel 1: key logits  key[b,a,s] = sum_c key_w[a,c]*x[b,c,s] + key_b[a]
// Writes into a [B, 16, HW] padded buffer; rows 4..15 hard-zeroed so the
// WMMA pooling kernel needs no mask. grid (HW/256, B), block 256.
// ---------------------------------------------------------------------------
__global__ void k_key_logits(const float* __restrict__ x,
                             const float* __restrict__ key_w,
                             const float* __restrict__ key_b,
                             float* __restrict__ keypad) {
    __shared__ float kw[A_ * C_];
    int tid = threadIdx.x;
    for (int i = tid; i < A_ * C_; i += 256) kw[i] = key_w[i];
    __syncthreads();

    int b = blockIdx.y;
    int s = blockIdx.x * 256 + tid;
    const float* xb = x + (size_t)b * C_ * HW_ + s;

    float a0 = 0.f, a1 = 0.f, a2 = 0.f, a3 = 0.f;
    for (int c = 0; c < C_; ++c) {
        float xv = xb[(size_t)c * HW_];
        if (c + 8 < C_) __builtin_prefetch(xb + (size_t)(c + 8) * HW_, 0, 0);
        a0 += kw[0 * C_ + c] * xv;
        a1 += kw[1 * C_ + c] * xv;
        a2 += kw[2 * C_ + c] * xv;
        a3 += kw[3 * C_ + c] * xv;
    }
    float* kb = keypad + (size_t)b * APAD_ * HW_ + s;
    kb[0 * HW_] = a0 + key_b[0];
    kb[1 * HW_] = a1 + key_b[1];
    kb[2 * HW_] = a2 + key_b[2];
    kb[3 * HW_] = a3 + key_b[3];
    #pragma unroll
    for (int a = A_; a < APAD_; ++a) kb[a * HW_] = 0.f;   // padding rows
}

// ---------------------------------------------------------------------------
// Kernel 2: in-place softmax over s for each of the B*A = 128 real rows.
// ---------------------------------------------------------------------------
__global__ void k_softmax(float* __restrict__ keypad) {
    __shared__ float sdata[256];
    int tid = threadIdx.x;
    int b = blockIdx.x >> 2;
    int a = blockIdx.x & 3;
    float* row = keypad + ((size_t)b * APAD_ + a) * HW_;

    float m = -INFINITY;
    for (int i = tid; i < HW_; i += 256) m = fmaxf(m, row[i]);
    sdata[tid] = m; __syncthreads();
    for (int st = 128; st > 0; st >>= 1) {
        if (tid < st) sdata[tid] = fmaxf(sdata[tid], sdata[tid + st]);
        __syncthreads();
    }
    m = sdata[0]; __syncthreads();

    float sum = 0.f;
    for (int i = tid; i < HW_; i += 256) {
        float e = __expf(row[i] - m);
        row[i] = e;
        sum += e;
    }
    sdata[tid] = sum; __syncthreads();
    for (int st = 128; st > 0; st >>= 1) {
        if (tid < st) sdata[tid] += sdata[tid + st];
        __syncthreads();
    }
    float inv = 1.f / sdata[0];
    for (int i = tid; i < HW_; i += 256) row[i] *= inv;
}

// ---------------------------------------------------------------------------
// Kernel 3: attn[b, c, a] = sum_s x[b,c,s] * key[b,a,s]   via WMMA f32 16x16x4
// B-matrix rows 4..15 are pre-zeroed padding -> inner loop is exactly
// two b64 loads + one v_wmma per K-step, EXEC all-1s throughout.
// grid 64 (2 blocks/batch), block 256 = 8 waves, one 16-channel tile per wave.
// ---------------------------------------------------------------------------
__global__ void k_attn_wmma(const float* __restrict__ x,
                            const float* __restrict__ keypad,
                            float* __restrict__ attnbuf) {
    int wave = threadIdx.x >> 5;
    int lane = threadIdx.x & 31;
    int b    = blockIdx.x >> 1;
    int tile = (blockIdx.x & 1) * 8 + wave;        // 0..15 channel tile

    int row = lane & 15;                           // M (A-frag) / N (B-frag)
    int kp  = lane >> 4;                           // K-pair select

    const float* xrow = x + (size_t)b * C_ * HW_
                          + (size_t)(tile * 16 + row) * HW_ + 2 * kp;
    const float* krow = keypad + ((size_t)b * APAD_ + row) * HW_ + 2 * kp;

    v8f acc = {0.f,0.f,0.f,0.f,0.f,0.f,0.f,0.f};
    for (int s = 0; s < HW_; s += 4) {
        v2f a  = *(const v2f*)(xrow + s);
        v2f bb = *(const v2f*)(krow + s);
        acc = __builtin_amdgcn_wmma_f32_16x16x4_f32(
                  false, a, false, bb, (short)0, acc, false, false);
    }
    if (row < A_) {
        for (int g = 0; g < 8; ++g) {
            int c = tile * 16 + g + 8 * kp;        // D row M = g + 8*kp
            attnbuf[(size_t)b * CA_ + (size_t)c * A_ + row] = acc[g];
        }
    }
}

// ---------------------------------------------------------------------------
// Kernel 4a: H[branch][b,m] = relu6( sum_i attn[b,i]*w1[m,i] + b1[m] )
// WMMA GEMM [32 x 1024] @ [1024 x 1024]^T ; 128 waves (2 branches x 64 m-tiles)
// ---------------------------------------------------------------------------
__global__ void k_mlp1(const float* __restrict__ attn,
                       const float* __restrict__ mul_w1,
                       const float* __restrict__ mul_b1,
                       const float* __restrict__ bias_w1,
                       const float* __restrict__ bias_b1,
                       float* __restrict__ Hbuf) {
    int wid  = blockIdx.x * 8 + (threadIdx.x >> 5);  // 0..127
    int lane = threadIdx.x & 31;
    int branch = wid >> 6;
    int mtile  = wid & 63;
    const float* w1 = branch ? bias_w1 : mul_w1;
    const float* b1 = branch ? bias_b1 : mul_b1;

    int row = lane & 15, kp = lane >> 4;
    const float* a0p = attn + (size_t)row * CA_ + 2 * kp;          // batches 0-15
    const float* a1p = attn + (size_t)(row + 16) * CA_ + 2 * kp;   // batches 16-31
    const float* bp  = w1 + (size_t)(mtile * 16 + row) * CA_ + 2 * kp;

    v8f acc0 = {0.f,0.f,0.f,0.f,0.f,0.f,0.f,0.f};
    v8f acc1 = {0.f,0.f,0.f,0.f,0.f,0.f,0.f,0.f};
    for (int i = 0; i < CA_; i += 4) {
        v2f a0 = *(const v2f*)(a0p + i);
        v2f a1 = *(const v2f*)(a1p + i);
        v2f bb = *(const v2f*)(bp + i);
        acc0 = __builtin_amdgcn_wmma_f32_16x16x4_f32(
                   false, a0, false, bb, (short)0, acc0, false, false);
        acc1 = __builtin_amdgcn_wmma_f32_16x16x4_f32(
                   false, a1, false, bb, (short)0, acc1, false, false);
    }
    int m = mtile * 16 + row;
    float bv = b1[m];
    float* Hb = Hbuf + (size_t)branch * B_ * MID_;
    for (int g = 0; g < 8; ++g) {
        int b0 = g + 8 * kp;
        Hb[(size_t)b0 * MID_ + m]        = fminf(fmaxf(acc0[g] + bv, 0.f), 6.f);
        Hb[(size_t)(b0 + 16) * MID_ + m] = fminf(fmaxf(acc1[g] + bv, 0.f), 6.f);
    }
}

// ---------------------------------------------------------------------------
// Kernel 4b: mul[b,o] = sigmoid(w2@H + b2)  /  bia[b,o] = w2@H + b2
// 32 waves (2 branches x 16 o-tiles), K = 1024.
// ---------------------------------------------------------------------------
__global__ void k_mlp2(const float* __restrict__ Hbuf,
                       const float* __restrict__ mul_w2,
                       const float* __restrict__ mul_b2,
                       const float* __restrict__ bias_w2,
                       const float* __restrict__ bias_b2,
                       float* __restrict__ mulbuf,
                       float* __restrict__ biabuf) {
    int wid  = blockIdx.x * 8 + (threadIdx.x >> 5);  // 0..31
    int lane = threadIdx.x & 31;
    int branch = wid >> 4;
    int otile  = wid & 15;
    const float* w2 = branch ? bias_w2 : mul_w2;
    const float* b2 = branch ? bias_b2 : mul_b2;
    const float* H  = Hbuf + (size_t)branch * B_ * MID_;

    int row = lane & 15, kp = lane >> 4;
    const float* a0p = H + (size_t)row * MID_ + 2 * kp;
    const float* a1p = H + (size_t)(row + 16) * MID_ + 2 * kp;
    const float* bp  = w2 + (size_t)(otile * 16 + row) * MID_ + 2 * kp;

    v8f acc0 = {0.f,0.f,0.f,0.f,0.f,0.f,0.f,0.f};
    v8f acc1 = {0.f,0.f,0.f,0.f,0.f,0.f,0.f,0.f};
    for (int i = 0; i < MID_; i += 4) {
        v2f a0 = *(const v2f*)(a0p + i);
        v2f a1 = *(const v2f*)(a1p + i);
        v2f bb = *(const v2f*)(bp + i);
        acc0 = __builtin_amdgcn_wmma_f32_16x16x4_f32(
                   false, a0, false, bb, (short)0, acc0, false, false);
        acc1 = __builtin_amdgcn_wmma_f32_16x16x4_f32(
                   false, a1, false, bb, (short)0, acc1, false, false);
    }
    int o = otile * 16 + row;
    float bv = b2[o];
    float* outp = branch ? biabuf : mulbuf;
    for (int g = 0; g < 8; ++g) {
        int b0 = g + 8 * kp;
        float v0 = acc0[g] + bv;
        float v1 = acc1[g] + bv;
        if (branch == 0) {
            v0 = 1.f / (1.f + __expf(-v0));
            v1 = 1.f / (1.f + __expf(-v1));
        }
        outp[(size_t)b0 * C_ + o]        = v0;
        outp[(size_t)(b0 + 16) * C_ + o] = v1;
    }
}

// ---------------------------------------------------------------------------
// Kernel 5: out = x * mul[b,c] + bia[b,c]  — pure bandwidth, b128 ld/st.
// Each 256-thread block covers 1024 consecutive floats -> uniform (b,c).
// ---------------------------------------------------------------------------
__global__ void k_apply(const float* __restrict__ x,
                        const float* __restrict__ mulbuf,
                        const float* __restrict__ biabuf,
                        float* __restrict__ out) {
    size_t i4   = (size_t)blockIdx.x * blockDim.x + threadIdx.x;
    size_t base = i4 * 4;
    int bc = (int)(base >> 12);                    // /HW_
    float m = mulbuf[bc];
    float a = biabuf[bc];
    v4f xv = *(const v4f*)(x + base);
    v4f r  = {xv.x * m + a, xv.y * m + a, xv.z * m + a, xv.w * m + a};
    *(v4f*)(out + base) = r;
}

// ---------------------------------------------------------------------------
extern "C" void kernel_launch(void* const* d_in, const int* in_sizes, int n_in,
                              void* d_out, int out_size, void* d_ws, size_t ws_size,
                              hipStream_t stream) {
    const float* x       = (const float*)d_in[0];
    const float* key_w   = (const float*)d_in[1];
    const float* key_b   = (const float*)d_in[2];
    const float* mul_w1  = (const float*)d_in[3];
    const float* mul_b1  = (const float*)d_in[4];
    const float* mul_w2  = (const float*)d_in[5];
    const float* mul_b2  = (const float*)d_in[6];
    const float* bias_w1 = (const float*)d_in[7];
    const float* bias_b1 = (const float*)d_in[8];
    const float* bias_w2 = (const float*)d_in[9];
    const float* bias_b2 = (const float*)d_in[10];
    float* out = (float*)d_out;

    float* ws      = (float*)d_ws;
    float* keypad  = ws;                                   // 32*16*4096 = 2097152
    float* attnbuf = keypad + (size_t)B_ * APAD_ * HW_;    // 32*1024   = 32768
    float* Hbuf    = attnbuf + (size_t)B_ * CA_;           // 2*32*1024 = 65536
    float* mulbuf  = Hbuf + 2 * (size_t)B_ * MID_;         // 32*256    = 8192
    float* biabuf  = mulbuf + (size_t)B_ * C_;             // 32*256    = 8192

    // 1. key logits into padded [B,16,HW] buffer (pad rows zeroed in-kernel)
    k_key_logits<<<dim3(HW_ / 256, B_), 256, 0, stream>>>(x, key_w, key_b, keypad);
    // 2. softmax over spatial dim (128 real rows)
    k_softmax<<<B_ * A_, 256, 0, stream>>>(keypad);
    // 3. softmax-weighted pooling (WMMA f32 16x16x4, maskless inner loop)
    k_attn_wmma<<<B_ * 2, 256, 0, stream>>>(x, keypad, attnbuf);
    // 4. SE-MLP, both branches (WMMA GEMMs)
    k_mlp1<<<16, 256, 0, stream>>>(attnbuf, mul_w1, mul_b1, bias_w1, bias_b1, Hbuf);
    k_mlp2<<<4, 256, 0, stream>>>(Hbuf, mul_w2, mul_b2, bias_w2, bias_b2, mulbuf, biabuf);
    // 5. elementwise scale + bias
    size_t n4 = (size_t)B_ * C_ * HW_ / 4;                 // 8388608
    k_apply<<<(unsigned)(n4 / 256), 256, 0, stream>>>(x, mulbuf, biabuf, out);
}